// VIT_46909632806985
// MI455X (gfx1250) — compile-verified
//
#include <hip/hip_runtime.h>

// CDNA5 / gfx1250 fused tiny-ViT forward.
// One wave32 per image; whole 4-block transformer fused in one kernel.
// ALL matmuls (QK^T, PV, QKV-proj, attn-proj, MLP FF1/FF2) run on
// V_WMMA_F32_16X16X4_F32 (native fp32 matrix op). VALU only does LN/softmax/
// patch-embed epilogues. Weight matrices are consumed as B-fragments loaded
// directly from global (L2-hot, identical across workgroups) -> small LDS.

typedef __attribute__((ext_vector_type(2))) float v2f;
typedef __attribute__((ext_vector_type(8))) float v8f;

#define T_TOK 50
#define T_PAD 64
#define NBLK 4
#define EPS_LN 1e-5f

#define WMMA4(a, b, c) \
  __builtin_amdgcn_wmma_f32_16x16x4_f32(false, (a), false, (b), (short)0, (c), false, false)

__device__ __forceinline__ void cpy32(float* dst, const float* src, int n, int lane) {
  for (int i = lane; i < n; i += 32) dst[i] = src[i];
}

__device__ __forceinline__ v2f ldg2(const float* p) { return *(const v2f*)p; }

__global__ __launch_bounds__(32) void vit_fused(
    const float* __restrict__ images,    // [B,1,28,28]
    const float* __restrict__ cls_tok,   // [1,8]
    const float* __restrict__ lin_W,     // [8,16]
    const float* __restrict__ ln1_g, const float* __restrict__ ln1_b,   // [4,8]
    const float* __restrict__ Wq, const float* __restrict__ Wk,
    const float* __restrict__ Wv,                                        // [4,8,8]
    const float* __restrict__ proj_W, const float* __restrict__ proj_b,  // [4,8,8],[4,8]
    const float* __restrict__ ln2_g, const float* __restrict__ ln2_b,    // [4,8]
    const float* __restrict__ W1, const float* __restrict__ b1,          // [4,32,8],[4,32]
    const float* __restrict__ W2, const float* __restrict__ b2,          // [4,8,32],[4,8]
    const float* __restrict__ mlp_W, const float* __restrict__ mlp_b,    // [10,8],[10]
    float* __restrict__ out)             // [B,10]
{
  // ---- LDS: state + small lane-indexed params only (~32 KB) ----
  __shared__ float w_lin[8 * 16];
  __shared__ float w_cls[8];
  __shared__ float w_l1g[NBLK][8], w_l1b[NBLK][8], w_l2g[NBLK][8], w_l2b[NBLK][8];
  __shared__ float w_pb[NBLK][8];
  __shared__ float w_b1[NBLK][32];
  __shared__ float w_b2[NBLK][8];
  __shared__ float w_mw[10][8];
  __shared__ float w_mb[10];
  __shared__ float posb[T_PAD][8];
  __shared__ float xb[T_PAD][9];   // residual stream x   (stride 9: bank pad)
  __shared__ float hb[T_PAD][9];   // LN1 out -> attn out (o) -> LN2 out (reused)
  __shared__ float qb[T_PAD][9], kb[T_PAD][9], vb[T_PAD][9];
  __shared__ float Sb[T_PAD][65]; // scores/probs (stride 65); reused as ff (cols<32)

  const int lane = threadIdx.x;
  const int hl = lane >> 4;        // half-wave id: WMMA K-split (K{0,1} vs K{2,3})
  const int l  = lane & 15;        // lane within half (M row / N col)

  // ---- small params into LDS ----
  cpy32(&w_lin[0],    lin_W,  128, lane);
  cpy32(&w_cls[0],    cls_tok,  8, lane);
  cpy32(&w_l1g[0][0], ln1_g,   32, lane);
  cpy32(&w_l1b[0][0], ln1_b,   32, lane);
  cpy32(&w_pb[0][0],  proj_b,  32, lane);
  cpy32(&w_l2g[0][0], ln2_g,   32, lane);
  cpy32(&w_l2b[0][0], ln2_b,   32, lane);
  cpy32(&w_b1[0][0],  b1,     128, lane);
  cpy32(&w_b2[0][0],  b2,      32, lane);
  cpy32(&w_mw[0][0],  mlp_W,   80, lane);
  cpy32(&w_mb[0],     mlp_b,   10, lane);

  // ---- sinusoidal positional table ----
  for (int idx = lane; idx < T_PAD * 8; idx += 32) {
    int t = idx >> 3, j = idx & 7;
    float val = 0.f;
    if (t < T_TOK) {
      float jj = (float)(j & ~1);
      float arg = (float)t * __powf(10000.f, -jj * 0.125f);
      val = (j & 1) ? __cosf(arg) : __sinf(arg);
    }
    posb[t][j] = val;
  }
  __syncthreads();

  // ---- patch embed + cls + pos (once per image; VALU) ----
  const float* img = images + (size_t)blockIdx.x * 784;
  for (int idx = lane; idx < T_PAD * 8; idx += 32) {
    int t = idx >> 3, d = idx & 7;
    float val = 0.f;
    if (t == 0) {
      val = w_cls[d] + posb[0][d];
    } else if (t < T_TOK) {
      int p = t - 1, pr = p / 7, pc = p % 7;
      float s = 0.f;
#pragma unroll
      for (int r = 0; r < 4; ++r)
#pragma unroll
        for (int c0 = 0; c0 < 4; ++c0)
          s += img[(pr * 4 + r) * 28 + pc * 4 + c0] * w_lin[d * 16 + r * 4 + c0];
      val = s + posb[t][d];
    }
    xb[t][d] = val;   // rows >= 50 stay zero (exact WMMA padding)
  }
  __syncthreads();

  const float scale = 0.3535533905932738f;  // n_embd^-0.5 (ref scales by C^-0.5)
  const int wrow8 = (l < 8) ? l : 7;        // clamped in-bounds weight row
  const float zm8 = (l < 8) ? 1.f : 0.f;    // zero-mask for N cols >= 8

  for (int i = 0; i < NBLK; ++i) {
    // ---- LN1 -> hb (padded rows zeroed) ----
    for (int t = lane; t < T_PAD; t += 32) {
      if (t < T_TOK) {
        float mu = 0.f;
#pragma unroll
        for (int d = 0; d < 8; ++d) mu += xb[t][d];
        mu *= 0.125f;
        float var = 0.f;
#pragma unroll
        for (int d = 0; d < 8; ++d) { float c = xb[t][d] - mu; var += c * c; }
        var *= 0.125f;
        float rs = rsqrtf(var + EPS_LN);
#pragma unroll
        for (int d = 0; d < 8; ++d)
          hb[t][d] = (xb[t][d] - mu) * rs * w_l1g[i][d] + w_l1b[i][d];
      } else {
#pragma unroll
        for (int d = 0; d < 8; ++d) hb[t][d] = 0.f;
      }
    }
    __syncthreads();

    // ---- QKV: [64,8] = hb[64,8] x W^T via WMMA (K=8 -> 2 k-steps) ----
    {
      v2f bq[2], bk[2], bv[2];
#pragma unroll
      for (int kt = 0; kt < 2; ++kt) {
        const int co = kt * 4 + 2 * hl;          // even -> 8B-aligned b64 loads
        bq[kt] = ldg2(Wq + i * 64 + wrow8 * 8 + co) * zm8;
        bk[kt] = ldg2(Wk + i * 64 + wrow8 * 8 + co) * zm8;
        bv[kt] = ldg2(Wv + i * 64 + wrow8 * 8 + co) * zm8;
      }
#pragma unroll
      for (int mt = 0; mt < 4; ++mt) {
        v2f a0, a1;
        a0.x = hb[mt * 16 + l][2 * hl + 0];
        a0.y = hb[mt * 16 + l][2 * hl + 1];
        a1.x = hb[mt * 16 + l][4 + 2 * hl + 0];
        a1.y = hb[mt * 16 + l][4 + 2 * hl + 1];
        v8f cq = {}, ck = {}, cv = {};
        cq = WMMA4(a0, bq[0], cq); cq = WMMA4(a1, bq[1], cq);
        ck = WMMA4(a0, bk[0], ck); ck = WMMA4(a1, bk[1], ck);
        cv = WMMA4(a0, bv[0], cv); cv = WMMA4(a1, bv[1], cv);
        if (l < 8) {
#pragma unroll
          for (int r = 0; r < 8; ++r) {
            int row = mt * 16 + r + 8 * hl;
            qb[row][l] = cq[r]; kb[row][l] = ck[r]; vb[row][l] = cv[r];
          }
        }
      }
    }
    __syncthreads();

    // ---- attention, one head at a time ----
    for (int h0 = 0; h0 < 2; ++h0) {
      // S = (Q*scale) x K^T  (4x4 tiles, K=4 exact)
      const int cq = 4 * h0 + 2 * hl;
      v2f bfr[4];
#pragma unroll
      for (int nt = 0; nt < 4; ++nt) {
        bfr[nt].x = kb[nt * 16 + l][cq + 0];
        bfr[nt].y = kb[nt * 16 + l][cq + 1];
      }
#pragma unroll
      for (int mt = 0; mt < 4; ++mt) {
        v2f a;
        a.x = qb[mt * 16 + l][cq + 0] * scale;
        a.y = qb[mt * 16 + l][cq + 1] * scale;
#pragma unroll
        for (int nt = 0; nt < 4; ++nt) {
          v8f c = {};
          c = WMMA4(a, bfr[nt], c);
#pragma unroll
          for (int r = 0; r < 8; ++r)
            Sb[mt * 16 + r + 8 * hl][nt * 16 + l] = c[r];
        }
      }
      __syncthreads();

      // row softmax over 50 valid cols; padded rows/cols zeroed
      for (int t = lane; t < T_PAD; t += 32) {
        if (t < T_TOK) {
          float m = -1e30f;
          for (int c0 = 0; c0 < T_TOK; ++c0) m = fmaxf(m, Sb[t][c0]);
          float s = 0.f;
          for (int c0 = 0; c0 < T_TOK; ++c0) {
            float e = __expf(Sb[t][c0] - m);
            Sb[t][c0] = e; s += e;
          }
          float inv = 1.f / s;
          for (int c0 = 0; c0 < T_TOK; ++c0) Sb[t][c0] *= inv;
          for (int c0 = T_TOK; c0 < T_PAD; ++c0) Sb[t][c0] = 0.f;
        } else {
          for (int c0 = 0; c0 < T_PAD; ++c0) Sb[t][c0] = 0.f;
        }
      }
      __syncthreads();

      // O = P x V_h  (K=64 -> 16 k-steps)
      const int vc = 4 * h0 + (l & 3);
      v8f acc[4] = {};
#pragma unroll
      for (int kt = 0; kt < 16; ++kt) {
        v2f b;
        b.x = (l < 4) ? vb[kt * 4 + 2 * hl + 0][vc] : 0.f;
        b.y = (l < 4) ? vb[kt * 4 + 2 * hl + 1][vc] : 0.f;
#pragma unroll
        for (int mt = 0; mt < 4; ++mt) {
          v2f a;
          a.x = Sb[mt * 16 + l][kt * 4 + 2 * hl + 0];
          a.y = Sb[mt * 16 + l][kt * 4 + 2 * hl + 1];
          acc[mt] = WMMA4(a, b, acc[mt]);
        }
      }
#pragma unroll
      for (int mt = 0; mt < 4; ++mt) {
        if (l < 4) {
#pragma unroll
          for (int r = 0; r < 8; ++r)
            hb[mt * 16 + r + 8 * hl][4 * h0 + l] = acc[mt][r];  // hb reused as o
        }
      }
      __syncthreads();
    }

    // ---- attn proj + bias + residual via WMMA ----
    {
      v2f bp[2];
#pragma unroll
      for (int kt = 0; kt < 2; ++kt)
        bp[kt] = ldg2(proj_W + i * 64 + wrow8 * 8 + kt * 4 + 2 * hl) * zm8;
#pragma unroll
      for (int mt = 0; mt < 4; ++mt) {
        v2f a0, a1;
        a0.x = hb[mt * 16 + l][2 * hl + 0];
        a0.y = hb[mt * 16 + l][2 * hl + 1];
        a1.x = hb[mt * 16 + l][4 + 2 * hl + 0];
        a1.y = hb[mt * 16 + l][4 + 2 * hl + 1];
        v8f c = {};
        c = WMMA4(a0, bp[0], c); c = WMMA4(a1, bp[1], c);
        if (l < 8) {
#pragma unroll
          for (int r = 0; r < 8; ++r) {
            int row = mt * 16 + r + 8 * hl;
            if (row < T_TOK) xb[row][l] += c[r] + w_pb[i][l];
          }
        }
      }
    }
    __syncthreads();

    // ---- LN2 -> hb ----
    for (int t = lane; t < T_PAD; t += 32) {
      if (t < T_TOK) {
        float mu = 0.f;
#pragma unroll
        for (int d = 0; d < 8; ++d) mu += xb[t][d];
        mu *= 0.125f;
        float var = 0.f;
#pragma unroll
        for (int d = 0; d < 8; ++d) { float c = xb[t][d] - mu; var += c * c; }
        var *= 0.125f;
        float rs = rsqrtf(var + EPS_LN);
#pragma unroll
        for (int d = 0; d < 8; ++d)
          hb[t][d] = (xb[t][d] - mu) * rs * w_l2g[i][d] + w_l2b[i][d];
      } else {
#pragma unroll
        for (int d = 0; d < 8; ++d) hb[t][d] = 0.f;
      }
    }
    __syncthreads();

    // ---- FF1: ff = relu(h2 x W1^T + b1) via WMMA (N=32: two dense N-tiles) ----
    {
      v2f bw[2][2];
#pragma unroll
      for (int nt = 0; nt < 2; ++nt)
#pragma unroll
        for (int kt = 0; kt < 2; ++kt)
          bw[nt][kt] = ldg2(W1 + i * 256 + (nt * 16 + l) * 8 + kt * 4 + 2 * hl);
#pragma unroll
      for (int mt = 0; mt < 4; ++mt) {
        v2f a0, a1;
        a0.x = hb[mt * 16 + l][2 * hl + 0];
        a0.y = hb[mt * 16 + l][2 * hl + 1];
        a1.x = hb[mt * 16 + l][4 + 2 * hl + 0];
        a1.y = hb[mt * 16 + l][4 + 2 * hl + 1];
#pragma unroll
        for (int nt = 0; nt < 2; ++nt) {
          v8f c = {};
          c = WMMA4(a0, bw[nt][0], c); c = WMMA4(a1, bw[nt][1], c);
#pragma unroll
          for (int r = 0; r < 8; ++r) {
            int row = mt * 16 + r + 8 * hl;
            float val = c[r] + w_b1[i][nt * 16 + l];
            Sb[row][nt * 16 + l] = (row < T_TOK) ? fmaxf(val, 0.f) : 0.f;
          }
        }
      }
    }
    __syncthreads();

    // ---- FF2: x += ff x W2^T + b2 via WMMA (K=32 -> 8 k-steps) ----
    {
      v2f b2f[8];
#pragma unroll
      for (int kt = 0; kt < 8; ++kt)
        b2f[kt] = ldg2(W2 + i * 256 + wrow8 * 32 + kt * 4 + 2 * hl) * zm8;
#pragma unroll
      for (int mt = 0; mt < 4; ++mt) {
        v8f c = {};
#pragma unroll
        for (int kt = 0; kt < 8; ++kt) {
          v2f a;
          a.x = Sb[mt * 16 + l][kt * 4 + 2 * hl + 0];
          a.y = Sb[mt * 16 + l][kt * 4 + 2 * hl + 1];
          c = WMMA4(a, b2f[kt], c);
        }
        if (l < 8) {
#pragma unroll
          for (int r = 0; r < 8; ++r) {
            int row = mt * 16 + r + 8 * hl;
            if (row < T_TOK) xb[row][l] += c[r] + w_b2[i][l];
          }
        }
      }
    }
    __syncthreads();
  }

  // ---- classifier head on cls token + softmax ----
  if (lane == 0) {
    float z[10];
    float m = -1e30f;
#pragma unroll
    for (int c0 = 0; c0 < 10; ++c0) {
      float s = w_mb[c0];
#pragma unroll
      for (int d = 0; d < 8; ++d) s += xb[0][d] * w_mw[c0][d];
      z[c0] = s; m = fmaxf(m, s);
    }
    float s = 0.f;
#pragma unroll
    for (int c0 = 0; c0 < 10; ++c0) { z[c0] = __expf(z[c0] - m); s += z[c0]; }
    float inv = 1.f / s;
#pragma unroll
    for (int c0 = 0; c0 < 10; ++c0)
      out[(size_t)blockIdx.x * 10 + c0] = z[c0] * inv;
  }
}

extern "C" void kernel_launch(void* const* d_in, const int* in_sizes, int n_in,
                              void* d_out, int out_size, void* d_ws, size_t ws_size,
                              hipStream_t stream) {
  (void)n_in; (void)out_size; (void)d_ws; (void)ws_size;
  const int B = in_sizes[0] / 784;
  vit_fused<<<dim3(B), dim3(32), 0, stream>>>(
      (const float*)d_in[0],  (const float*)d_in[1],  (const float*)d_in[2],
      (const float*)d_in[3],  (const float*)d_in[4],  (const float*)d_in[5],
      (const float*)d_in[6],  (const float*)d_in[7],  (const float*)d_in[8],
      (const float*)d_in[9],  (const float*)d_in[10], (const float*)d_in[11],
      (const float*)d_in[12], (const float*)d_in[13], (const float*)d_in[14],
      (const float*)d_in[15], (const float*)d_in[16], (const float*)d_in[17],
      (float*)d_out);
}